// TransportOperator_11699490915107
// MI455X (gfx1250) — compile-verified
//
#include <hip/hip_runtime.h>
#include <hip/hip_bf16.h>
#include <math.h>

// ---------------------------------------------------------------------------
// TransportOperator loss on MI455X (gfx1250).
//  recon: per-batch 64x64 expm via scaling-squaring + order-12 Taylor Horner,
//         all matmuls on V_WMMA_F32_16X16X4_F32 (f32 tensor path, wave32).
//  One workgroup (512 thr = 16 waves) per batch element; whole problem in LDS.
// ---------------------------------------------------------------------------

typedef __attribute__((ext_vector_type(2))) float v2f;
typedef __attribute__((ext_vector_type(8))) float v8f;

#define LDA 66           // padded row stride (floats) for 64x64 LDS matrices
#define TAYLOR_N 12      // ||As|| <= 0.5 -> remainder ~0.5^13/13! ~ 2e-14

// D = scale * (Amat @ Bmat) + diag_add * I     (all 64x64 f32 in LDS)
// wave w owns output tile (w>>2, w&3); 16 chained wmma f32 16x16x4 per tile.
__device__ __forceinline__ void mm64_wmma(const float* __restrict__ Amat,
                                          const float* __restrict__ Bmat,
                                          float* __restrict__ Dmat,
                                          float scale, float diag_add,
                                          int wave, int lane) {
  const int tm   = wave >> 2;      // output row tile 0..3
  const int tn   = wave & 3;       // output col tile 0..3
  const int half = lane >> 4;      // 0: lanes 0-15, 1: lanes 16-31
  const int l16  = lane & 15;

  v8f acc = {0.f, 0.f, 0.f, 0.f, 0.f, 0.f, 0.f, 0.f};

  const int arow = (tm * 16 + l16) * LDA;   // A fragment: per-lane row
  const int bcol = tn * 16 + l16;           // B fragment: per-lane col

#pragma unroll
  for (int k0 = 0; k0 < 64; k0 += 4) {
    // A 16x4: VGPR0 = K {k0 | k0+2}, VGPR1 = K {k0+1 | k0+3} by lane half
    v2f a, b;
    a.x = Amat[arow + k0 + 2 * half];
    a.y = Amat[arow + k0 + 2 * half + 1];
    // B 4x16: row-striped across lanes; VGPR r holds rows {r, r+2}
    b.x = Bmat[(k0 + 2 * half) * LDA + bcol];
    b.y = Bmat[(k0 + 2 * half + 1) * LDA + bcol];
    acc = __builtin_amdgcn_wmma_f32_16x16x4_f32(
        /*neg_a=*/false, a, /*neg_b=*/false, b,
        /*c_mod=*/(short)0, acc, /*reuse_a=*/false, /*reuse_b=*/false);
  }

  // C/D 16x16 f32: VGPR v -> rows {tm*16+v | tm*16+8+v} by lane half
#pragma unroll
  for (int v = 0; v < 8; ++v) {
    const int r  = tm * 16 + half * 8 + v;
    const int cc = tn * 16 + l16;
    float val = acc[v] * scale;
    if (r == cc) val += diag_add;
    Dmat[r * LDA + cc] = val;
  }
}

__global__ __launch_bounds__(512)
void expm_recon_kernel(const float* __restrict__ z0,
                       const float* __restrict__ z1,
                       const float* __restrict__ c,
                       const float* __restrict__ psi,
                       float* __restrict__ out, int B) {
  const int b    = blockIdx.x;
  const int tid  = threadIdx.x;
  const int lane = tid & 31;
  const int wave = tid >> 5;     // 0..15

  __shared__ float M0[64 * LDA];   // As (scaled generator)
  __shared__ float M1[64 * LDA];   // ping
  __shared__ float M2[64 * LDA];   // pong
  __shared__ float cs[16];
  __shared__ float red[64];
  __shared__ float zin[64];
  __shared__ float sScale;
  __shared__ int   sCount;

  if (tid < 16) cs[tid]  = c[b * 16 + tid];
  if (tid < 64) zin[tid] = z0[(size_t)b * 64 + tid];
  __syncthreads();

  // ---- A[j,k] = sum_m c[b,m] * psi[j,k,m]  (psi hot in 192MB L2) ----
#pragma unroll
  for (int e = 0; e < 8; ++e) {
    const int idx = tid + e * 512;          // 0..4095
    const int j = idx >> 6, k = idx & 63;
    const float* __restrict__ p = psi + (size_t)idx * 16;
    float acc = 0.f;
#pragma unroll
    for (int m = 0; m < 16; ++m) acc = fmaf(cs[m], p[m], acc);
    M0[j * LDA + k] = acc;
  }
  __syncthreads();

  // ---- inf-norm -> squaring count s, scale A by 2^-s ----
  if (tid < 64) {
    float s = 0.f;
#pragma unroll 4
    for (int k = 0; k < 64; ++k) s += fabsf(M0[tid * LDA + k]);
    red[tid] = s;
  }
  __syncthreads();
  if (tid == 0) {
    float mx = 0.f;
    for (int i = 0; i < 64; ++i) mx = fmaxf(mx, red[i]);
    int s = 0;
    float t = mx;
    while (t > 0.5f && s < 24) { t *= 0.5f; ++s; }   // uniform scalar loop
    sCount = s;
    sScale = exp2f(-(float)s);
  }
  __syncthreads();
  const int   nsq    = sCount;
  const float scaleA = sScale;
#pragma unroll
  for (int e = 0; e < 8; ++e) {
    const int idx = tid + e * 512;
    const int j = idx >> 6, k = idx & 63;
    M0[j * LDA + k] *= scaleA;
  }
  __syncthreads();

  // ---- Horner Taylor: P = I + As/1 (I + As/2 ( ... (I + As/N) )) ----
  // init P = I + As/N
#pragma unroll
  for (int e = 0; e < 8; ++e) {
    const int idx = tid + e * 512;
    const int j = idx >> 6, k = idx & 63;
    M1[j * LDA + k] = M0[j * LDA + k] * (1.0f / TAYLOR_N) +
                      ((j == k) ? 1.0f : 0.0f);
  }
  __syncthreads();

  float* P = M1;
  float* T = M2;
  for (int k = TAYLOR_N - 1; k >= 1; --k) {
    mm64_wmma(M0, P, T, 1.0f / (float)k, 1.0f, wave, lane);  // T = As@P/k + I
    __syncthreads();
    float* tmp = P; P = T; T = tmp;
  }

  // ---- squaring phase: E <- E@E, nsq times (uniform trip count) ----
  for (int i = 0; i < nsq; ++i) {
    mm64_wmma(P, P, T, 1.0f, 0.0f, wave, lane);
    __syncthreads();
    float* tmp = P; P = T; T = tmp;
  }

  // ---- w = E @ z0[b]; accumulate sum (z1 - w)^2 / B ----
  if (tid < 64) {
    float w = 0.f;
#pragma unroll 4
    for (int k = 0; k < 64; ++k) w = fmaf(P[tid * LDA + k], zin[k], w);
    const float d = z1[(size_t)b * 64 + tid] - w;
    red[tid] = d * d;
  }
  __syncthreads();
  if (tid == 0) {
    float s = 0.f;
    for (int i = 0; i < 64; ++i) s += red[i];
    atomicAdd(out, s / (float)B);
  }
}

// gamma * sum(psi^2) + zeta * sum(|c|)  (tiny, bandwidth-trivial)
__global__ __launch_bounds__(256)
void reg_kernel(const float* __restrict__ psi, int npsi,
                const float* __restrict__ c, int nc,
                float* __restrict__ out) {
  const int stride = gridDim.x * blockDim.x;
  float acc = 0.f;
  for (int i = blockIdx.x * blockDim.x + threadIdx.x; i < npsi; i += stride) {
    const float v = psi[i];
    acc = fmaf(0.01f * v, v, acc);
  }
  for (int i = blockIdx.x * blockDim.x + threadIdx.x; i < nc; i += stride) {
    acc += 0.01f * fabsf(c[i]);
  }
  __shared__ float sm[256];
  sm[threadIdx.x] = acc;
  __syncthreads();
  for (int off = 128; off > 0; off >>= 1) {
    if (threadIdx.x < off) sm[threadIdx.x] += sm[threadIdx.x + off];
    __syncthreads();
  }
  if (threadIdx.x == 0) atomicAdd(out, sm[0]);
}

__global__ void init_out_kernel(float* out) { out[0] = 0.0f; }

extern "C" void kernel_launch(void* const* d_in, const int* in_sizes, int n_in,
                              void* d_out, int out_size, void* d_ws, size_t ws_size,
                              hipStream_t stream) {
  const float* z0  = (const float*)d_in[0];   // [B,64]
  const float* z1  = (const float*)d_in[1];   // [B,64]
  const float* c   = (const float*)d_in[2];   // [B,1,16]
  const float* psi = (const float*)d_in[3];   // [64,64,16]
  float* out = (float*)d_out;

  const int B = in_sizes[0] / 64;             // 8192

  init_out_kernel<<<1, 1, 0, stream>>>(out);
  reg_kernel<<<64, 256, 0, stream>>>(psi, in_sizes[3], c, in_sizes[2], out);
  expm_recon_kernel<<<B, 512, 0, stream>>>(z0, z1, c, psi, out, B);
}